// OurModel_69483980914783
// MI455X (gfx1250) — compile-verified
//
#include <hip/hip_runtime.h>
#include <hip/hip_bf16.h>

typedef __attribute__((ext_vector_type(2))) float v2f;
typedef __attribute__((ext_vector_type(8))) float v8f;

// B=128, D=256, H=8, DH=32, FFN=2048, HID=64, K=3

// ---------------------------------------------------------------------------
// Factorized MHA + residual + LayerNorm.
// att[b,h,q,k] = Q[aq]*Kv[ak]*G[fq,fk]  (q = aq*8+fq, k = ak*8+fk)
// softmax over q (axis=2);  out[b,h*32+m] = sum_{ff,ki} S[m*8+ff, ki]*W[ki,ff]
// One block per (b, dir); 256 threads.
// ---------------------------------------------------------------------------
__global__ void mha_kernel(const float* __restrict__ image, const float* __restrict__ text,
                           const float* __restrict__ wq, const float* __restrict__ wk,
                           const float* __restrict__ wv, const float* __restrict__ wf,
                           const float* __restrict__ g, const float* __restrict__ bb,
                           float* __restrict__ oA) {
  const int b = blockIdx.x;
  const int dir = blockIdx.y;
  const float* qin  = dir ? text  : image;
  const float* kvin = dir ? image : text;
  const float* Wq = wq + dir * 256;
  const float* Wk = wk + dir * 256;
  const float* Wv = wv + dir * 256;
  const float* Wf = wf + dir * 256;
  const int t = threadIdx.x;

  __shared__ float sQ[256], sKv[256];
  __shared__ float sG[64], sW[64];
  __shared__ float sDen[256];
  __shared__ float sS[2048];
  __shared__ float sOut[256];
  __shared__ float red[256];

  sQ[t]  = qin[b * 256 + t];
  sKv[t] = kvin[b * 256 + t];
  if (t < 64) {
    int i = t >> 3, j = t & 7;
    float sg = 0.f, sw = 0.f;
    for (int dh = 0; dh < 32; ++dh) {
      sg += Wq[i * 32 + dh] * Wk[j * 32 + dh];
      sw += Wv[i * 32 + dh] * Wf[j * 32 + dh];
    }
    sG[t] = 0.5f * sg;  // G[fq][fk], SCALE folded in
    sW[t] = sw;         // W[fk][ff]
  }
  __syncthreads();

  for (int h = 0; h < 8; ++h) {
    // pass 1: thread = (kk, ki); denominator over all q
    {
      int kk = t >> 3, ki = t & 7;
      float Kvv = sKv[h * 32 + kk];
      float d = 0.f;
      for (int qq = 0; qq < 32; ++qq) {
        float x = sQ[h * 32 + qq] * Kvv;
        #pragma unroll
        for (int qi = 0; qi < 8; ++qi) d += __expf(x * sG[qi * 8 + ki]);
      }
      sDen[t] = 1.0f / d;
    }
    __syncthreads();
    // pass 2: thread = q = (qq*8+qi); S[q,ki] = sum_kk P*Kv[kk]
    {
      int qq = t >> 3, qi = t & 7;
      float Qv = sQ[h * 32 + qq];
      float s[8];
      #pragma unroll
      for (int ki = 0; ki < 8; ++ki) s[ki] = 0.f;
      for (int kk = 0; kk < 32; ++kk) {
        float Kvv = sKv[h * 32 + kk];
        float x = Qv * Kvv;
        #pragma unroll
        for (int ki = 0; ki < 8; ++ki)
          s[ki] += __expf(x * sG[qi * 8 + ki]) * sDen[kk * 8 + ki] * Kvv;
      }
      #pragma unroll
      for (int ki = 0; ki < 8; ++ki) sS[t * 8 + ki] = s[ki];
    }
    __syncthreads();
    if (t < 32) {
      float o = 0.f;
      #pragma unroll
      for (int ff = 0; ff < 8; ++ff)
        #pragma unroll
        for (int ki = 0; ki < 8; ++ki)
          o += sS[(t * 8 + ff) * 8 + ki] * sW[ki * 8 + ff];
      sOut[h * 32 + t] = sQ[h * 32 + t] + o;  // residual q_in + attn out
    }
    __syncthreads();
  }

  // LayerNorm over the 256-row
  float v = sOut[t];
  red[t] = v; __syncthreads();
  for (int s = 128; s > 0; s >>= 1) { if (t < s) red[t] += red[t + s]; __syncthreads(); }
  float mu = red[0] * (1.0f / 256.0f); __syncthreads();
  float d0 = v - mu;
  red[t] = d0 * d0; __syncthreads();
  for (int s = 128; s > 0; s >>= 1) { if (t < s) red[t] += red[t + s]; __syncthreads(); }
  float var = red[0] * (1.0f / 256.0f);
  oA[(dir * 128 + b) * 256 + t] = d0 * rsqrtf(var + 1e-5f) * g[dir * 256 + t] + bb[dir * 256 + t];
}

// ---------------------------------------------------------------------------
// WMMA fp32 GEMM: C[M,N] = act(A[M,K] @ W[N,K]^T + bias[N])
// One wave computes a 16x64 band of C (4 N-tiles), so the A fragment is loaded
// once per k-step and reused by 4 v_wmma_f32_16x16x4_f32 (5 b64 loads / 4 wmma).
// A frag (16x4): lane<16 -> {A[r][k0],A[r][k0+1]}, lane>=16 -> {A[r][k0+2],A[r][k0+3]}
// B frag (4x16): same K split; B[k][n] = W[n][k] -> contiguous pair in W's row.
// C frag: vgpr r = C[tm*16 + (lane<16 ? r : r+8)][tn*16 + lane%16]
// Requires N % 64 == 0, K % 4 == 0, M % 16 == 0 (true for all call sites).
// ---------------------------------------------------------------------------
__global__ void gemm_xwt(const float* __restrict__ A, const float* __restrict__ W,
                         const float* __restrict__ bias, float* __restrict__ C,
                         int M, int N, int K, int relu) {
  const int groupsN = N >> 6;                       // 4 tiles of 16 per wave
  const int groups = (M >> 4) * groupsN;
  const int group = blockIdx.x * (blockDim.x >> 5) + (threadIdx.x >> 5);
  if (group >= groups) return;
  const int tm = group / groupsN;
  const int tn0 = (group - tm * groupsN) * 4;       // first N-tile index
  const int lane = threadIdx.x & 31;
  const int hi = lane >> 4;
  const int l = lane & 15;
  const int c = hi * 2;

  const float* Arow = A + (size_t)(tm * 16 + l) * K + c;
  const float* W0 = W + (size_t)((tn0 + 0) * 16 + l) * K + c;
  const float* W1 = W + (size_t)((tn0 + 1) * 16 + l) * K + c;
  const float* W2 = W + (size_t)((tn0 + 2) * 16 + l) * K + c;
  const float* W3 = W + (size_t)((tn0 + 3) * 16 + l) * K + c;

  v8f acc0 = {0.f, 0.f, 0.f, 0.f, 0.f, 0.f, 0.f, 0.f};
  v8f acc1 = acc0, acc2 = acc0, acc3 = acc0;

  #pragma unroll 2
  for (int k0 = 0; k0 < K; k0 += 4) {
    v2f a, b0, b1, b2, b3;
    a.x  = Arow[k0]; a.y  = Arow[k0 + 1];
    b0.x = W0[k0];   b0.y = W0[k0 + 1];
    b1.x = W1[k0];   b1.y = W1[k0 + 1];
    b2.x = W2[k0];   b2.y = W2[k0 + 1];
    b3.x = W3[k0];   b3.y = W3[k0 + 1];
    acc0 = __builtin_amdgcn_wmma_f32_16x16x4_f32(false, a, false, b0, (short)0, acc0, false, false);
    acc1 = __builtin_amdgcn_wmma_f32_16x16x4_f32(false, a, false, b1, (short)0, acc1, false, false);
    acc2 = __builtin_amdgcn_wmma_f32_16x16x4_f32(false, a, false, b2, (short)0, acc2, false, false);
    acc3 = __builtin_amdgcn_wmma_f32_16x16x4_f32(false, a, false, b3, (short)0, acc3, false, false);
  }

  const int rbase = tm * 16 + hi * 8;
  v8f accs[4] = {acc0, acc1, acc2, acc3};
  #pragma unroll
  for (int j = 0; j < 4; ++j) {
    const int col = (tn0 + j) * 16 + l;
    const float bv = bias ? bias[col] : 0.0f;
    #pragma unroll
    for (int r = 0; r < 8; ++r) {
      float v = accs[j][r] + bv;
      if (relu) v = fmaxf(v, 0.0f);
      C[(size_t)(rbase + r) * N + col] = v;
    }
  }
}

// residual + LayerNorm over 256; writes into dst with stride/offset (for concat)
__global__ void resln_kernel(const float* __restrict__ x, const float* __restrict__ tmp,
                             const float* __restrict__ g, const float* __restrict__ bb,
                             float* __restrict__ dst, int dstStride, int dstOff) {
  const int b = blockIdx.x, t = threadIdx.x;
  __shared__ float red[256];
  float v = x[b * 256 + t] + tmp[b * 256 + t];
  red[t] = v; __syncthreads();
  for (int s = 128; s > 0; s >>= 1) { if (t < s) red[t] += red[t + s]; __syncthreads(); }
  float mu = red[0] * (1.0f / 256.0f); __syncthreads();
  float d = v - mu;
  red[t] = d * d; __syncthreads();
  for (int s = 128; s > 0; s >>= 1) { if (t < s) red[t] += red[t + s]; __syncthreads(); }
  float var = red[0] * (1.0f / 256.0f);
  dst[b * dstStride + dstOff + t] = d * rsqrtf(var + 1e-5f) * g[t] + bb[t];
}

// z = zm + eps * exp(0.5 * zlv)
__global__ void z_kernel(const float* __restrict__ zm, const float* __restrict__ zlv,
                         const float* __restrict__ eps, float* __restrict__ z, int n) {
  int i = blockIdx.x * blockDim.x + threadIdx.x;
  if (i < n) z[i] = zm[i] + eps[i] * __expf(0.5f * zlv[i]);
}

// per (k, half): mean + covariance (correction=1) of 64x64 slice, LDS-staged
__global__ void cov_kernel(const float* __restrict__ zbuf, float* __restrict__ mus,
                           float* __restrict__ covs) {
  const int bx = blockIdx.x;       // 0..5 : k = bx/2, half = bx%2
  const int k = bx >> 1, half = bx & 1;
  const float* Z = zbuf + k * 128 * 64 + half * 64 * 64;
  __shared__ float sz[64 * 64];
  __shared__ float smu[64];
  const int t = threadIdx.x;
  for (int i = t; i < 4096; i += 256) sz[i] = Z[i];
  __syncthreads();
  if (t < 64) {
    float s = 0.f;
    for (int r = 0; r < 64; ++r) s += sz[r * 64 + t];
    s *= (1.0f / 64.0f);
    smu[t] = s;
    mus[bx * 64 + t] = s;
  }
  __syncthreads();
  for (int i = t; i < 4096; i += 256) sz[i] -= smu[i & 63];
  __syncthreads();
  float* C = covs + bx * 4096;
  for (int e = t; e < 4096; e += 256) {
    int i = e >> 6, j = e & 63;
    float s = 0.f;
    for (int r = 0; r < 64; ++r) s += sz[r * 64 + i] * sz[r * 64 + j];
    C[e] = s * (1.0f / 63.0f);
  }
}

// mixture over k: mu = mean_k mus; sig = mean_k(cov + dev dev^T)
__global__ void mix_kernel(const float* __restrict__ mus, const float* __restrict__ covs,
                           float* __restrict__ out) {
  const int cls = blockIdx.x, t = threadIdx.x;
  __shared__ float smu[64], sdev[3 * 64];
  if (t < 64) {
    float m0 = mus[(0 * 2 + cls) * 64 + t];
    float m1 = mus[(1 * 2 + cls) * 64 + t];
    float m2 = mus[(2 * 2 + cls) * 64 + t];
    float mm = (m0 + m1 + m2) * (1.0f / 3.0f);
    smu[t] = mm; sdev[t] = m0 - mm; sdev[64 + t] = m1 - mm; sdev[128 + t] = m2 - mm;
  }
  __syncthreads();
  float* base = out + 4 * 128 * 256 + cls * (64 + 4096);
  if (t < 64) base[t] = smu[t];
  float* sig = base + 64;
  for (int e = t; e < 4096; e += 256) {
    int i = e >> 6, j = e & 63;
    float s = 0.f;
    #pragma unroll
    for (int k = 0; k < 3; ++k)
      s += covs[(k * 2 + cls) * 4096 + e] + sdev[k * 64 + i] * sdev[k * 64 + j];
    sig[e] = s * (1.0f / 3.0f);
  }
}

extern "C" void kernel_launch(void* const* d_in, const int* in_sizes, int n_in,
                              void* d_out, int out_size, void* d_ws, size_t ws_size,
                              hipStream_t stream) {
  const float* image    = (const float*)d_in[0];
  const float* text     = (const float*)d_in[1];
  const float* wq       = (const float*)d_in[2];
  const float* wk       = (const float*)d_in[3];
  const float* wv       = (const float*)d_in[4];
  const float* wf       = (const float*)d_in[5];
  const float* ln_a_g   = (const float*)d_in[6];
  const float* ln_a_b   = (const float*)d_in[7];
  const float* ffn_w1   = (const float*)d_in[8];
  const float* ffn_b1   = (const float*)d_in[9];
  const float* ffn_w2   = (const float*)d_in[10];
  const float* ffn_b2   = (const float*)d_in[11];
  const float* ln_f_g   = (const float*)d_in[12];
  const float* ln_f_b   = (const float*)d_in[13];
  const float* fusion_w = (const float*)d_in[14];
  const float* fusion_b = (const float*)d_in[15];
  const float* ew1      = (const float*)d_in[16];
  const float* eb1      = (const float*)d_in[17];
  const float* ew2      = (const float*)d_in[18];
  const float* eb2      = (const float*)d_in[19];
  const float* ewm      = (const float*)d_in[20];
  const float* ebm      = (const float*)d_in[21];
  const float* ewv      = (const float*)d_in[22];
  const float* ebv      = (const float*)d_in[23];
  const float* ewd1     = (const float*)d_in[24];
  const float* ebd1     = (const float*)d_in[25];
  const float* ewd2     = (const float*)d_in[26];
  const float* ebd2     = (const float*)d_in[27];
  const float* eps      = (const float*)d_in[28];
  float* out = (float*)d_out;

  float* ws     = (float*)d_ws;
  float* oA     = ws;               // 2*128*256   = 65536
  float* hbuf   = oA + 65536;       // 2*128*2048  = 524288
  float* tmp2   = hbuf + 524288;    // 2*128*256   = 65536
  float* concat = tmp2 + 65536;     // 128*512     = 65536
  float* eh1    = concat + 65536;   // 128*64      = 8192
  float* eh     = eh1 + 8192;       // 8192
  float* ezm    = eh + 8192;        // 8192
  float* ezlv   = ezm + 8192;       // 8192
  float* zbuf   = ezlv + 8192;      // 3*128*64    = 24576
  float* er1    = zbuf + 24576;     // 128*256     = 32768
  float* musb   = er1 + 32768;      // 3*2*64      = 384
  float* covb   = musb + 384;       // 3*2*4096    = 24576

  auto gemm = [&](const float* A, const float* Wp, const float* bp, float* Cp,
                  int M, int N, int K, int relu) {
    int groups = (M / 16) * (N / 64);             // one wave per 16x64 band
    gemm_xwt<<<(groups + 7) / 8, 256, 0, stream>>>(A, Wp, bp, Cp, M, N, K, relu);
  };

  // fused multimodal attention (both directions), residual + ln_a
  mha_kernel<<<dim3(128, 2), 256, 0, stream>>>(image, text, wq, wk, wv, wf,
                                               ln_a_g, ln_a_b, oA);

  // FFN per direction -> concat buffer (row stride 512)
  for (int dir = 0; dir < 2; ++dir) {
    const float* x = oA + dir * 32768;
    gemm(x, ffn_w1 + dir * 2048 * 256, ffn_b1 + dir * 2048,
         hbuf + dir * 262144, 128, 2048, 256, 1);
    gemm(hbuf + dir * 262144, ffn_w2 + dir * 256 * 2048, ffn_b2 + dir * 256,
         tmp2 + dir * 32768, 128, 256, 2048, 0);
    resln_kernel<<<128, 256, 0, stream>>>(x, tmp2 + dir * 32768,
                                          ln_f_g + dir * 256, ln_f_b + dir * 256,
                                          concat, 512, dir * 256);
  }

  // fusion: (128,512) @ (256,512)^T + b -> fused (first output section)
  gemm(concat, fusion_w, fusion_b, out, 128, 256, 512, 0);

  // experts: fused, image, text
  for (int k = 0; k < 3; ++k) {
    const float* xk = (k == 0) ? out : (k == 1 ? image : text);
    gemm(xk, ew1 + k * 64 * 256, eb1 + k * 64, eh1, 128, 64, 256, 1);
    gemm(eh1, ew2 + k * 64 * 64, eb2 + k * 64, eh, 128, 64, 64, 0);
    gemm(eh, ewm + k * 64 * 64, ebm + k * 64, ezm, 128, 64, 64, 0);
    gemm(eh, ewv + k * 64 * 64, ebv + k * 64, ezlv, 128, 64, 64, 0);
    z_kernel<<<32, 256, 0, stream>>>(ezm, ezlv, eps + k * 8192, zbuf + k * 8192, 8192);
    gemm(zbuf + k * 8192, ewd1 + k * 256 * 64, ebd1 + k * 256, er1, 128, 256, 64, 1);
    gemm(er1, ewd2 + k * 256 * 256, ebd2 + k * 256, out + (1 + k) * 32768, 128, 256, 256, 0);
  }

  // per-class Gaussian fit + 3-component mixture
  cov_kernel<<<6, 256, 0, stream>>>(zbuf, musb, covb);
  mix_kernel<<<2, 256, 0, stream>>>(musb, covb, out);

  (void)in_sizes; (void)n_in; (void)out_size; (void)ws_size;
}